// RotaryPositionalEmbedding_21328807592314
// MI455X (gfx1250) — compile-verified
//
#include <hip/hip_runtime.h>
#include <math.h>
#include <stdint.h>

// RoPE for x[4,32,4096,128] fp32. Memory-bound: 512MB traffic @ 23.3 TB/s ~ 22us.
// Strategy:
//   kernel 1: build cos/sin table [seq, 64] float2 into d_ws (256K sincos instead of 33.5M).
//   kernel 2: tiled apply; per block async-stage 4KB of table into LDS (CDNA5
//             global_load_async_to_lds_b128), keep per-thread table float4 in
//             registers, stream x/out with 128-bit non-temporal accesses,
//             reusing the table value across 16 batch*head rows.

typedef float v4f __attribute__((ext_vector_type(4)));

#define ROPE_THETA 10000.0f

// ---------------- kernel 1: trig table ----------------
__global__ void rope_build_table(const long long* __restrict__ tpos,
                                 float2* __restrict__ tbl,
                                 int seq, int half) {
  int id = blockIdx.x * blockDim.x + threadIdx.x;
  int total = seq * half;
  if (id >= total) return;
  int p = id / half;
  int j = id - p * half;
  // inv_freq = theta^(-j/half); precise powf (only 256K evaluations total)
  float inv = powf(ROPE_THETA, -(float)j / (float)half);
  float ang = (float)tpos[p] * inv;
  // precise sinf/cosf: angles up to ~4095 rad need real range reduction
  tbl[id] = make_float2(cosf(ang), sinf(ang));
}

// ---------------- kernel 2: tiled apply ----------------
// block: 256 threads = 8 positions x 32 float4-lanes; loops over 16 bh rows.
// grid: (seq/8, bh/16)
__global__ __launch_bounds__(256)
void rope_apply_tiled(const float* __restrict__ x,
                      float* __restrict__ out,
                      const v4f* __restrict__ tbl4,   // [seq*32] float4 = (c0,s0,c1,s1)
                      int seq) {
  __shared__ v4f lds_tbl[256];

  const int u = threadIdx.x;
  const int P = blockIdx.x * 8;    // position-tile base
  const int B = blockIdx.y * 16;   // batch*head tile base

  // --- CDNA5 async global->LDS stage of the 4KB table tile (rows P..P+7) ---
  {
    uint32_t lds_off = (uint32_t)(uintptr_t)(&lds_tbl[u]);
    uint64_t gaddr   = (uint64_t)(uintptr_t)(tbl4 + (size_t)P * 32 + (size_t)u);
    asm volatile("global_load_async_to_lds_b128 %0, %1, off"
                 :: "v"(lds_off), "v"(gaddr)
                 : "memory");
    asm volatile("s_wait_asynccnt 0" ::: "memory");
  }
  __syncthreads();

  // each thread's (c0,s0,c1,s1) -> registers, reused for all 16 rows
  const v4f t = lds_tbl[u];        // ds_load_b128

  const int p = u >> 5;            // position within tile (0..7)
  const int q = u & 31;            // float4 lane within the 128-wide row
  const size_t row_stride = (size_t)seq * 128;
  const size_t base = ((size_t)B * seq + (size_t)(P + p)) * 128 + (size_t)q * 4;

#pragma unroll
  for (int i = 0; i < 16; ++i) {
    size_t off = base + (size_t)i * row_stride;
    v4f v = __builtin_nontemporal_load((const v4f*)(x + off));
    v4f r;
    r.x = t.x * v.x - t.y * v.y;   // pair j0: cos*e - sin*o
    r.y = t.y * v.x + t.x * v.y;   //          sin*e + cos*o
    r.z = t.z * v.z - t.w * v.w;   // pair j1
    r.w = t.w * v.z + t.z * v.w;
    __builtin_nontemporal_store(r, (v4f*)(out + off));
  }
}

// ---------------- fallback: generic, inline sincos ----------------
__global__ void rope_apply_fallback(const float* __restrict__ x,
                                    float* __restrict__ out,
                                    const long long* __restrict__ tpos,
                                    int seq, long long n4) {
  long long idx = (long long)blockIdx.x * blockDim.x + threadIdx.x;
  if (idx >= n4) return;
  int q = (int)(idx & 31);
  long long row = idx >> 5;
  int pos = (int)(row % (long long)seq);
  float fp = (float)tpos[pos];
  int j0 = 2 * q;
  float a0 = fp * powf(ROPE_THETA, -(float)(j0)     / 64.0f);
  float a1 = fp * powf(ROPE_THETA, -(float)(j0 + 1) / 64.0f);
  float c0 = cosf(a0), s0 = sinf(a0);
  float c1 = cosf(a1), s1 = sinf(a1);
  size_t off = (size_t)idx * 4;
  v4f v = *(const v4f*)(x + off);
  v4f r;
  r.x = c0 * v.x - s0 * v.y;
  r.y = s0 * v.x + c0 * v.y;
  r.z = c1 * v.z - s1 * v.w;
  r.w = s1 * v.z + c1 * v.w;
  *(v4f*)(out + off) = r;
}

extern "C" void kernel_launch(void* const* d_in, const int* in_sizes, int n_in,
                              void* d_out, int out_size, void* d_ws, size_t ws_size,
                              hipStream_t stream) {
  const float* x = (const float*)d_in[0];
  const long long* tpos = (const long long*)d_in[1];
  float* out = (float*)d_out;

  const long long n = (long long)in_sizes[0];   // total elements of x
  const int seq = in_sizes[1];                  // token_positions length
  const int D = 128, HALF = 64;
  const long long rows = n / D;
  const long long bh = (seq > 0) ? rows / (long long)seq : 0;
  const long long n4 = n / 4;

  const size_t tbl_bytes = (size_t)seq * HALF * sizeof(float2);
  const bool tiled_ok =
      (ws_size >= tbl_bytes) &&
      (seq % 8 == 0) && (bh > 0) && (bh % 16 == 0) &&
      (n == bh * (long long)seq * D);

  if (tiled_ok) {
    int tthreads = seq * HALF;
    rope_build_table<<<(tthreads + 255) / 256, 256, 0, stream>>>(
        tpos, (float2*)d_ws, seq, HALF);
    dim3 grid((unsigned)(seq / 8), (unsigned)(bh / 16));
    rope_apply_tiled<<<grid, 256, 0, stream>>>(
        x, out, (const v4f*)d_ws, seq);
  } else {
    rope_apply_fallback<<<(unsigned)((n4 + 255) / 256), 256, 0, stream>>>(
        x, out, tpos, seq, n4);
  }
}